// CNN_Encoder_60945585930895
// MI455X (gfx1250) — compile-verified
//
#include <hip/hip_runtime.h>

// ---------------------------------------------------------------------------
// CNN encoder == GEMM:  out[m,k] = relu( sum_c A[m,c] * Weff[c,k] + b[k] )
//   A[m,c]    = user_batch[(b*2048 + l)*128 + c],  m = b*2040 + l  (M=65280)
//   Weff[c,k] = filt_flat[c] * W_k[c,k]            (K=1024, N=256)
// fp32 path -> V_WMMA_F32_16X16X4_F32 (CDNA5 native fp32 matrix op).
// Double-buffered LDS pipeline fed by GLOBAL_LOAD_ASYNC_TO_LDS_B128
// (ASYNCcnt) when available; register pipeline fallback otherwise.
// Weff is pre-packed in global memory as K-pairs so B fragments are single
// aligned ds_load_b64 and all staging is contiguous b128 copies.
// ---------------------------------------------------------------------------

typedef float v2f __attribute__((ext_vector_type(2)));
typedef float v8f __attribute__((ext_vector_type(8)));
typedef int   v4i __attribute__((ext_vector_type(4)));

#if defined(__gfx1250__) && __has_builtin(__builtin_amdgcn_global_load_async_to_lds_b128)
#define USE_ASYNC 1
#else
#define USE_ASYNC 0
#endif

#if defined(__gfx1250__) && __has_builtin(__builtin_amdgcn_s_wait_asynccnt)
#define WAIT_ASYNC() __builtin_amdgcn_s_wait_asynccnt(0)
#else
#define WAIT_ASYNC() asm volatile("s_wait_asynccnt 0x0" ::: "memory")
#endif

// exact operand types per hipcc diagnostic: int4* in AS1 (global) / AS3 (LDS)
typedef __attribute__((address_space(1))) v4i g_v4i;
typedef __attribute__((address_space(3))) v4i l_v4i;

#define IN_D   128
#define KDIM   1024          // F*D
#define NOUT   256
#define LSEQ   2048
#define NWIN   2040
#define BATCH  32
#define MTOT   (BATCH * NWIN)   // 65280, divisible by 128

#define MBLK   128
#define NBLK   128
#define KBLK   32
#define ASTR   36               // padded A LDS stride (floats)
#define NPAIR  (KBLK / 2)       // 16 K-pairs per slice
#define BSTR   288              // padded B pair stride (floats): +288 == +32 banks

// ---------------------------------------------------------------------------
// Kernel 1: fold filt into W_k AND pre-pack into K-pair layout:
//   WeffP[(c/2)*256*2 + k*2 + (c&1)] = filt[c] * W_k[c*256 + k]
// so {Weff[2p][n], Weff[2p+1][n]} are adjacent -> b128 staging, b64 fragments.
// ---------------------------------------------------------------------------
__global__ __launch_bounds__(256)
void weff_kernel(const float* __restrict__ filt,
                 const float* __restrict__ Wk,
                 float* __restrict__ WeffP) {
    const int c = blockIdx.x;        // 0..1023 (f*128+d)
    const int k = threadIdx.x;       // 0..255
    WeffP[((c >> 1) * NOUT + k) * 2 + (c & 1)] = filt[c] * Wk[c * NOUT + k];
}

// ---------------------------------------------------------------------------
// Kernel 2: tiled WMMA GEMM + bias + relu
//   grid = (510, 2), block = 256 threads = 8 wave32s (2 waves/SIMD)
//   wave grid 4(M) x 2(N); wave tile 32x64 = 2x4 WMMA 16x16 accumulators
// ---------------------------------------------------------------------------
__global__ __launch_bounds__(256, 2)
void gemm_kernel(const float* __restrict__ U,      // user_batch flat
                 const float* __restrict__ WeffP,  // 512 pairs x 256 cols x 2
                 const float* __restrict__ bias,   // 256
                 float* __restrict__ out) {        // 65280 x 256
    __shared__ float AL[2][MBLK * ASTR];           // 2 x 18 KB
    __shared__ float BL[2][NPAIR * BSTR];          // 2 x 18 KB, K-pair packed

    const int tid  = threadIdx.x;
    const int lane = tid & 31;
    const int w    = tid >> 5;          // wave 0..7
    const int wm   = w & 3;             // wave M index 0..3
    const int wn   = w >> 2;            // wave N index 0..1
    const int mblk = blockIdx.x * MBLK;
    const int nblk = blockIdx.y * NBLK;

    // ---- staging slot descriptors (float offsets), 4 x 16B per matrix ----
    int aGlob[4], aLds[4];              // A tile: 128 rows x 32 cols
    int bGlob[4], bLds[4];              // B tile: 16 pair-rows x 256 floats
#pragma unroll
    for (int i = 0; i < 4; ++i) {
        const int s    = tid + i * 256;       // 0..1023 float4 slots
        // A: 8 float4 per 32-col row
        const int arow = s >> 3;
        const int akc  = (s & 7) << 2;
        const int m    = mblk + arow;
        const int b    = m / NWIN;            // batch-boundary rows handled here
        const int l    = m - b * NWIN;
        aGlob[i] = (b * LSEQ + l) * IN_D + akc;      // + kk -> element
        aLds[i]  = arow * ASTR + akc;
        // B: 64 float4 per pair-row (128 cols x 2 packed floats)
        const int bpr  = s >> 6;
        const int bcq  = (s & 63) << 2;
        bGlob[i] = bpr * (NOUT * 2) + nblk * 2 + bcq;  // + kk*NOUT -> element
        bLds[i]  = bpr * BSTR + bcq;
    }

    v8f acc[2][4];
#pragma unroll
    for (int mt = 0; mt < 2; ++mt)
#pragma unroll
        for (int nt = 0; nt < 4; ++nt)
            acc[mt][nt] = (v8f){0.f, 0.f, 0.f, 0.f, 0.f, 0.f, 0.f, 0.f};

#if USE_ASYNC
    // ---- async staging: global -> LDS with no VGPR data (ASYNCcnt) ----
    auto issueAsync = [&](int kk, int bsel) {
#pragma unroll
        for (int i = 0; i < 4; ++i)
            __builtin_amdgcn_global_load_async_to_lds_b128(
                (g_v4i*)(U + aGlob[i] + kk),
                (l_v4i*)(&AL[bsel][aLds[i]]), 0, 0);
#pragma unroll
        for (int i = 0; i < 4; ++i)
            __builtin_amdgcn_global_load_async_to_lds_b128(
                (g_v4i*)(WeffP + bGlob[i] + kk * NOUT),
                (l_v4i*)(&BL[bsel][bLds[i]]), 0, 0);
    };
#else
    float4 aReg[4], bReg[4];
    auto loadGlobal = [&](int kk) {
#pragma unroll
        for (int i = 0; i < 4; ++i)
            aReg[i] = *(const float4*)(U + aGlob[i] + kk);
#pragma unroll
        for (int i = 0; i < 4; ++i)
            bReg[i] = *(const float4*)(WeffP + bGlob[i] + kk * NOUT);
    };
    auto storeLds = [&](int bsel) {
#pragma unroll
        for (int i = 0; i < 4; ++i)
            *(float4*)(&AL[bsel][aLds[i]]) = aReg[i];
#pragma unroll
        for (int i = 0; i < 4; ++i)
            *(float4*)(&BL[bsel][bLds[i]]) = bReg[i];
    };
#endif

    auto compute = [&](int bsel) {
#pragma unroll
        for (int ks = 0; ks < KBLK; ks += 4) {
            // fragment layout: lanes 0-15 hold K=ks,ks+1; lanes 16-31 K=ks+2,ks+3
            const int kf = ks + ((lane >> 4) << 1);
            const int pr = (ks >> 1) + (lane >> 4);
            const int nr = lane & 15;
            v2f af[2];
#pragma unroll
            for (int mt = 0; mt < 2; ++mt)
                af[mt] = *(const v2f*)(&AL[bsel][(wm * 32 + mt * 16 + nr) * ASTR + kf]);
            v2f bf[4];
#pragma unroll
            for (int nt = 0; nt < 4; ++nt)
                bf[nt] = *(const v2f*)(&BL[bsel][pr * BSTR + (wn * 64 + nt * 16 + nr) * 2]);
#pragma unroll
            for (int mt = 0; mt < 2; ++mt)
#pragma unroll
                for (int nt = 0; nt < 4; ++nt)
                    acc[mt][nt] = __builtin_amdgcn_wmma_f32_16x16x4_f32(
                        false, af[mt], false, bf[nt],
                        (short)0, acc[mt][nt], false, false);
        }
    };

    // ---- double-buffered pipeline: copies for slice s+1 run under compute(s) ----
#if USE_ASYNC
    issueAsync(0, 0);
    WAIT_ASYNC();
    __syncthreads();
    int bsel = 0;
    for (int kk = 0; kk < KDIM; kk += KBLK) {
        const bool more = (kk + KBLK) < KDIM;
        if (more) issueAsync(kk + KBLK, bsel ^ 1);   // DMA hides under WMMAs
        compute(bsel);
        if (more) WAIT_ASYNC();
        __syncthreads();
        bsel ^= 1;
    }
#else
    loadGlobal(0);
    storeLds(0);
    __syncthreads();
    int bsel = 0;
    for (int kk = 0; kk < KDIM; kk += KBLK) {
        const bool more = (kk + KBLK) < KDIM;
        if (more) loadGlobal(kk + KBLK);             // VMEM hides under WMMAs
        compute(bsel);
        if (more) storeLds(bsel ^ 1);
        __syncthreads();
        bsel ^= 1;
    }
#endif

    // ---- epilogue: bias + relu ----
    // C/D layout: VGPR r <-> M=r (lanes 0-15) / M=r+8 (lanes 16-31); lane&15 <-> N
#pragma unroll
    for (int mt = 0; mt < 2; ++mt) {
#pragma unroll
        for (int nt = 0; nt < 4; ++nt) {
            const int gn = nblk + wn * 64 + nt * 16 + (lane & 15);
            const float bv = bias[gn];
            const int gmBase = mblk + wm * 32 + mt * 16 + ((lane >> 4) << 3);
#pragma unroll
            for (int r = 0; r < 8; ++r) {
                const float v = acc[mt][nt][r] + bv;
                out[(long)(gmBase + r) * NOUT + gn] = v > 0.f ? v : 0.f;
            }
        }
    }
}

// ---------------------------------------------------------------------------
extern "C" void kernel_launch(void* const* d_in, const int* in_sizes, int n_in,
                              void* d_out, int out_size, void* d_ws, size_t ws_size,
                              hipStream_t stream) {
    (void)in_sizes; (void)n_in; (void)out_size; (void)ws_size;
    const float* U    = (const float*)d_in[0];   // (32, 2048, 128)
    const float* filt = (const float*)d_in[1];   // (8, 128) -> flat 1024
    const float* Wk   = (const float*)d_in[2];   // (1024, 256)
    const float* bk   = (const float*)d_in[3];   // (256,)
    float* WeffP = (float*)d_ws;                 // 1 MB scratch, pair-packed
    float* out   = (float*)d_out;                // (32, 2040, 256)

    weff_kernel<<<dim3(KDIM), 256, 0, stream>>>(filt, Wk, WeffP);
    gemm_kernel<<<dim3(MTOT / MBLK, NOUT / NBLK), 256, 0, stream>>>(U, WeffP, bk, out);
}